// Router_6116033429797
// MI455X (gfx1250) — compile-verified
//
#include <hip/hip_runtime.h>
#include <stdint.h>

typedef float v2f __attribute__((ext_vector_type(2)));
typedef float v8f __attribute__((ext_vector_type(8)));
typedef unsigned int v4u __attribute__((ext_vector_type(4)));
typedef int v4i __attribute__((ext_vector_type(4)));
typedef int v8i __attribute__((ext_vector_type(8)));

#define HDIM   2048
#define NEXP   64
#define NTOK   16384
#define SEQ    4096
#define NBATCH 4
#define KCHUNK 256                 // K columns of W staged per LDS buffer
#define NCHUNK (HDIM / KCHUNK)     // 8
#define LDS_STRIDE (KCHUNK + 4)    // pad 4 dwords: bank = (4*e + col) % 64, conflict-free
#define ALPHA_F 0.001f
#define EPS_F   1e-20f
#define NEG_INF (-3.402823466e38f)

#ifdef __has_builtin
#if __has_builtin(__builtin_amdgcn_tensor_load_to_lds)
#define HAVE_TDM 1
#endif
#endif
#ifndef HAVE_TDM
#define HAVE_TDM 0
#endif

// ---------------------------------------------------------------------------
// TDM: DMA one 64 x KCHUNK f32 tile of W into LDS (with 4-dword row padding).
// D# group0: count=1, lds_addr, global_addr(57b), type=2.
// D# group1: data_size=4B, pad_enable, pad_interval=256dw, pad_amount=4dw,
//            tensor_dim0=2048, tensor_dim1=64, tile_dim0=KCHUNK, tile_dim1=64,
//            tensor_dim0_stride=2048.
// ---------------------------------------------------------------------------
__device__ __forceinline__ void stage_w_tdm(const float* Wg, uint32_t lds_off,
                                            int chunk) {
#if HAVE_TDM
    uint64_t ga = (uint64_t)(uintptr_t)Wg + (uint64_t)chunk * (KCHUNK * 4);
    v4u g0;
    g0[0] = 1u;                                        // count=1, user mode
    g0[1] = lds_off;                                   // lds_addr (bytes)
    g0[2] = (uint32_t)(ga & 0xffffffffu);              // global_addr[31:0]
    g0[3] = (uint32_t)((ga >> 32) & 0x01ffffffu)       // global_addr[56:32]
          | (2u << 30);                                // type = 2 ("image")
    v8i g1;
    g1[0] = (int)((2u << 16)                           // data_size = 4B
                | (1u << 20)                           // pad_enable
                | (7u << 22)                           // pad_interval: 256 dwords
                | (3u << 25));                         // pad_amount: 4 dwords
    g1[1] = (int)((uint32_t)(HDIM & 0xffff) << 16);    // tensor_dim0[15:0]
    g1[2] = (int)(((uint32_t)NEXP << 16)               // tensor_dim1[15:0]
                | ((uint32_t)HDIM >> 16));             // tensor_dim0[31:16]
    g1[3] = (int)((uint32_t)KCHUNK << 16);             // tile_dim0
    g1[4] = (int)NEXP;                                 // tile_dim1 (tile_dim2=0)
    g1[5] = (int)HDIM;                                 // tensor_dim0_stride[31:0]
    g1[6] = 0;
    g1[7] = 0;
    v4i z4 = {0, 0, 0, 0};
#if __clang_major__ >= 23
    v8i z8 = {0, 0, 0, 0, 0, 0, 0, 0};
    __builtin_amdgcn_tensor_load_to_lds(g0, g1, z4, z4, z8, 0);
#else
    __builtin_amdgcn_tensor_load_to_lds(g0, g1, z4, z4, 0);
#endif
#endif
}

// Fallback: cooperative global->LDS staging (global_load_b128 + ds_store_b32).
__device__ __forceinline__ void stage_w_coop(float* dst, const float* Wg,
                                             int chunk, int tid) {
#if !HAVE_TDM
    const int nvec = NEXP * KCHUNK / 4;
    for (int i = tid; i < nvec; i += 128) {
        const int idx = i * 4;
        const int row = idx / KCHUNK;
        const int col = idx % KCHUNK;
        const float4 v =
            *(const float4*)(Wg + (long)row * HDIM + chunk * KCHUNK + col);
        float* d = dst + row * LDS_STRIDE + col;
        d[0] = v.x; d[1] = v.y; d[2] = v.z; d[3] = v.w;
    }
#endif
}

// ---------------------------------------------------------------------------
// Kernel A: fused gate GEMM (WMMA f32 16x16x4) + softmax + top-2 + aux stats.
// One wave = 32 tokens x 64 experts (8 accumulators). Block = 4 waves.
// W K-chunks double-buffered in LDS via TDM; x streamed from global.
// ---------------------------------------------------------------------------
__global__ __launch_bounds__(128) void router_gemm_topk(
    const float* __restrict__ x,      // [16384, 2048]
    const float* __restrict__ W,      // [64, 2048]
    float* __restrict__ out_idx,      // [16384, 2] (indices as float)
    float* __restrict__ out_w,        // [16384, 2]
    float* __restrict__ ws_ssum,      // [4, 64]
    float* __restrict__ ws_ce)        // [4, 64]
{
    __shared__ float smw[2][NEXP * LDS_STRIDE];

    const int tid   = threadIdx.x;
    const int lane  = tid & 31;
    const int wave  = tid >> 5;           // 0..3
    const int lmod  = lane & 15;
    const int half  = lane >> 4;          // A/B: K sub-offset; C/D: row +8
    const int tokenBase = blockIdx.x * 128 + wave * 32;
    const int bIdx  = (blockIdx.x * 128) >> 12;   // uniform per block

    const float* ap0 = x + (long)(tokenBase + lmod) * HDIM + 2 * half;
    const float* ap1 = x + (long)(tokenBase + 16 + lmod) * HDIM + 2 * half;

    v8f acc[2][4] = {};

    // Prologue: stage chunk 0 into buffer 0.
    if (HAVE_TDM) {
        if (wave == 0)
            stage_w_tdm(W, (uint32_t)(uintptr_t)&smw[0][0], 0);
    } else {
        stage_w_coop(&smw[0][0], W, 0, tid);
    }
#if HAVE_TDM
    if (wave == 0) __builtin_amdgcn_s_wait_tensorcnt(0);
#endif
    __syncthreads();

    for (int c = 0; c < NCHUNK; ++c) {
        const int buf = c & 1;
        // Kick off the next chunk's DMA while we compute this one.
        if (c + 1 < NCHUNK) {
            if (HAVE_TDM) {
                if (wave == 0)
                    stage_w_tdm(W, (uint32_t)(uintptr_t)&smw[buf ^ 1][0], c + 1);
            } else {
                stage_w_coop(&smw[buf ^ 1][0], W, c + 1, tid);
            }
        }

        const float* bw = &smw[buf][0];
        const int kbase = c * KCHUNK;
        for (int kk0 = 0; kk0 < KCHUNK; kk0 += 64) {
            __builtin_prefetch(ap0 + kbase + kk0 + 512, 0, 3);
            __builtin_prefetch(ap1 + kbase + kk0 + 512, 0, 3);
#pragma unroll
            for (int kk = 0; kk < 64; kk += 4) {
                const int kg = kbase + kk0 + kk;   // global K
                const int kl = kk0 + kk;           // K within chunk
                v2f a0 = *(const v2f*)(ap0 + kg);
                v2f a1 = *(const v2f*)(ap1 + kg);
#pragma unroll
                for (int t = 0; t < 4; ++t) {
                    v2f b = *(const v2f*)(bw + (16 * t + lmod) * LDS_STRIDE +
                                          kl + 2 * half);
                    acc[0][t] = __builtin_amdgcn_wmma_f32_16x16x4_f32(
                        false, a0, false, b, (short)0, acc[0][t], false, false);
                    acc[1][t] = __builtin_amdgcn_wmma_f32_16x16x4_f32(
                        false, a1, false, b, (short)0, acc[1][t], false, false);
                }
            }
        }

#if HAVE_TDM
        if (wave == 0 && c + 1 < NCHUNK) __builtin_amdgcn_s_wait_tensorcnt(0);
#endif
        __syncthreads();
    }

    // ---- Fused epilogue ----------------------------------------------------
    // C/D layout: reg r, lanes 0-15 -> (row r, expert lmod+16t),
    //             lanes 16-31 -> (row r+8, expert lmod+16t)
    float scacc0 = 0.f, scacc1 = 0.f, scacc2 = 0.f, scacc3 = 0.f;

#pragma unroll
    for (int g = 0; g < 2; ++g) {
#pragma unroll
        for (int r = 0; r < 8; ++r) {
            float v0 = acc[g][0][r], v1 = acc[g][1][r];
            float v2 = acc[g][2][r], v3 = acc[g][3][r];

            float m = fmaxf(fmaxf(v0, v1), fmaxf(v2, v3));
#pragma unroll
            for (int off = 1; off < 16; off <<= 1)
                m = fmaxf(m, __shfl_xor(m, off, 32));

            float p0 = __expf(v0 - m), p1 = __expf(v1 - m);
            float p2 = __expf(v2 - m), p3 = __expf(v3 - m);
            float s = p0 + p1 + p2 + p3;
#pragma unroll
            for (int off = 1; off < 16; off <<= 1)
                s += __shfl_xor(s, off, 32);
            float inv = 1.0f / s;

            scacc0 += p0 * inv; scacc1 += p1 * inv;
            scacc2 += p2 * inv; scacc3 += p3 * inv;

            // top-2 on logits (softmax monotone); tie-break = lower index
            float t0v = NEG_INF, t1v = NEG_INF;
            int   t0i = -1,      t1i = -1;
            auto ins = [&](float cv, int ci) {
                if (cv > t0v || (cv == t0v && ci < t0i)) {
                    t1v = t0v; t1i = t0i; t0v = cv; t0i = ci;
                } else if (cv > t1v || (cv == t1v && ci < t1i)) {
                    t1v = cv; t1i = ci;
                }
            };
            ins(v0, lmod);      ins(v1, lmod + 16);
            ins(v2, lmod + 32); ins(v3, lmod + 48);
#pragma unroll
            for (int off = 1; off < 16; off <<= 1) {
                float ov0 = __shfl_xor(t0v, off, 32);
                int   oi0 = __shfl_xor(t0i, off, 32);
                float ov1 = __shfl_xor(t1v, off, 32);
                int   oi1 = __shfl_xor(t1i, off, 32);
                ins(ov0, oi0);
                ins(ov1, oi1);
            }

            if (lmod == 0) {
                const int row = tokenBase + g * 16 + r + 8 * half;
                float q0 = __expf(t0v - m) * inv;
                float q1 = __expf(t1v - m) * inv;
                float rec = 1.0f / (q0 + q1 + EPS_F);
                out_idx[2 * row + 0] = (float)t0i;
                out_idx[2 * row + 1] = (float)t1i;
                out_w[2 * row + 0] = q0 * rec;
                out_w[2 * row + 1] = q1 * rec;
                atomicAdd(&ws_ce[bIdx * NEXP + t0i], 1.0f);
                atomicAdd(&ws_ce[bIdx * NEXP + t1i], 1.0f);
            }
        }
    }

    // combine halves (same experts) before the global atomic
    scacc0 += __shfl_xor(scacc0, 16, 32);
    scacc1 += __shfl_xor(scacc1, 16, 32);
    scacc2 += __shfl_xor(scacc2, 16, 32);
    scacc3 += __shfl_xor(scacc3, 16, 32);
    if (half == 0) {
        atomicAdd(&ws_ssum[bIdx * NEXP + lmod +  0], scacc0);
        atomicAdd(&ws_ssum[bIdx * NEXP + lmod + 16], scacc1);
        atomicAdd(&ws_ssum[bIdx * NEXP + lmod + 32], scacc2);
        atomicAdd(&ws_ssum[bIdx * NEXP + lmod + 48], scacc3);
    }
}

// ---------------------------------------------------------------------------
// Kernel B: fold per-(batch,expert) stats into the scalar aux loss.
// ---------------------------------------------------------------------------
__global__ __launch_bounds__(256) void router_aux(
    const float* __restrict__ ws_ssum,
    const float* __restrict__ ws_ce,
    float* __restrict__ out_aux)
{
    __shared__ float red[256];
    const int t = threadIdx.x;                    // t = b*64 + e
    const float ce_norm = ws_ce[t] * ((float)NEXP / (float)(SEQ * 2));
    const float mean_sc = ws_ssum[t] * (1.0f / (float)SEQ);
    red[t] = ce_norm * mean_sc;
    __syncthreads();
#pragma unroll
    for (int s = 128; s > 0; s >>= 1) {
        if (t < s) red[t] += red[t + s];
        __syncthreads();
    }
    if (t == 0) out_aux[0] = red[0] * (ALPHA_F / (float)NBATCH);
}

extern "C" void kernel_launch(void* const* d_in, const int* in_sizes, int n_in,
                              void* d_out, int out_size, void* d_ws, size_t ws_size,
                              hipStream_t stream) {
    const float* x = (const float*)d_in[0];   // [4,4096,2048]
    const float* W = (const float*)d_in[1];   // [64,2048]
    float* out      = (float*)d_out;
    float* out_idx  = out;                    // 16384*2
    float* out_w    = out + NTOK * 2;         // 16384*2
    float* out_aux  = out + NTOK * 4;         // 1
    float* ws       = (float*)d_ws;
    float* ws_ssum  = ws;                     // 256 floats
    float* ws_ce    = ws + NBATCH * NEXP;     // 256 floats

    hipMemsetAsync(d_ws, 0, 2 * NBATCH * NEXP * sizeof(float), stream);
    router_gemm_topk<<<NTOK / 128, 128, 0, stream>>>(x, W, out_idx, out_w,
                                                     ws_ssum, ws_ce);
    router_aux<<<1, 256, 0, stream>>>(ws_ssum, ws_ce, out_aux);
}